// IE_62500364091609
// MI455X (gfx1250) — compile-verified
//
#include <hip/hip_runtime.h>
#include <math.h>

#define NBITS  12
#define SFULL  4096     // padded lattice (masks 0..4095); column j <-> mask j+1
#define NMASKS 4095     // valid masks 1..4095
#define NVARSC 4094     // ie_vars count (masks 1..4094)
#define BROWS  4096

typedef __attribute__((ext_vector_type(2))) float v2f;
typedef __attribute__((ext_vector_type(8))) float v8f;

// ---------------------------------------------------------------------------
// Kernel A (1 block): fuzzy-measure lattice DP (level-by-level over popcount)
// followed by in-place fast Mobius transform, all in LDS. Emits padded w[4096]
// (w[j] = Mobius coeff for mask j+1, w[4095] = 0 pad).
// ---------------------------------------------------------------------------
__global__ __launch_bounds__(256)
void IE_fm_mobius_kernel(const float* __restrict__ ie_vars,
                         float* __restrict__ wpad) {
  __shared__ float a[SFULL];
  const int tid = threadIdx.x;
  const int nt  = blockDim.x;

  // fm[m] = |v[m]| + max over immediate sub-subsets (previous popcount level)
  for (int k = 1; k <= NBITS - 1; ++k) {
    for (int m = 1 + tid; m <= NVARSC; m += nt) {
      if (__popc(m) == k) {
        float val = fabsf(ie_vars[m - 1]);
        if (k > 1) {
          float mx = 0.0f;
#pragma unroll
          for (int b = 0; b < NBITS; ++b)
            if (m & (1 << b)) mx = fmaxf(mx, a[m ^ (1 << b)]);
          val += mx;
        }
        a[m] = val;
      }
    }
    __syncthreads();
  }

  // a[0]=0 (empty set), a[4095]=1, clamp rest to 1.0 (matches reference order)
  for (int m = tid; m < SFULL; m += nt) {
    if (m == 0)           a[m] = 0.0f;
    else if (m == NMASKS) a[m] = 1.0f;
    else                  a[m] = fminf(a[m], 1.0f);
  }
  __syncthreads();

  // Fast Mobius: w[m] = sum_{s subset m} (-1)^{|m|-|s|} fm[s].
  // Per bit pass: masks with bit set are written, masks without are only read.
  for (int b = 0; b < NBITS; ++b) {
    for (int m = tid; m < SFULL; m += nt) {
      if (m & (1 << b)) a[m] -= a[m ^ (1 << b)];
    }
    __syncthreads();
  }

  for (int j = tid; j < SFULL; j += nt)
    wpad[j] = (j < NMASKS) ? a[j + 1] : 0.0f;
}

// ---------------------------------------------------------------------------
// Kernel B: one wave (32 lanes) per 16-row tile.
// Fused: logsum tile via V_WMMA_F32_16X16X4_F32 (K=12 -> 3 chained WMMAs,
// x2 for lo/hi), exp (v_exp_f32), weight by w from LDS, column reduction via
// lane shuffles. Never materializes the 4096x4095 subset-product matrix.
// ---------------------------------------------------------------------------
__global__ __launch_bounds__(32)
void IE_choquet_kernel(const float* __restrict__ x,
                       const float* __restrict__ wpad,
                       float* __restrict__ out) {
  __shared__ float ws[SFULL];
  const int lane = threadIdx.x;          // 0..31, wave32
  for (int j = lane; j < SFULL; j += 32) ws[j] = wpad[j];
  __syncthreads();

  const int rowbase = blockIdx.x * 16;
  const int half    = lane >> 4;         // 0: K even pair low, 1: K pair high
  const int lm      = lane & 15;         // A: row M ; B/C: column N
  const float* xr   = x + (rowbase + lm) * (2 * NBITS);

  // A fragments (16x4 f32): lane -> M=l%16, component v -> K = 4j + 2*half + v
  v2f al[3], au[3];
#pragma unroll
  for (int j = 0; j < 3; ++j) {
    const int kb = 4 * j + 2 * half;
    al[j].x = logf(fmaxf(xr[kb],             1e-20f));
    al[j].y = logf(fmaxf(xr[kb + 1],         1e-20f));
    au[j].x = logf(fmaxf(xr[NBITS + kb],     1e-20f));
    au[j].y = logf(fmaxf(xr[NBITS + kb + 1], 1e-20f));
  }

  float accl[8] = {};
  float accu[8] = {};

  for (int c = 0; c < 256; ++c) {
    const int mask = 16 * c + lm + 1;    // column -> lattice mask (pad col: w=0)
    v8f cl = {};
    v8f cu = {};
#pragma unroll
    for (int j = 0; j < 3; ++j) {
      const int kb = 4 * j + 2 * half;
      v2f bf;
      bf.x = (float)((mask >> kb) & 1);
      bf.y = (float)((mask >> (kb + 1)) & 1);
      cl = __builtin_amdgcn_wmma_f32_16x16x4_f32(false, al[j], false, bf,
                                                 (short)0, cl, false, false);
      cu = __builtin_amdgcn_wmma_f32_16x16x4_f32(false, au[j], false, bf,
                                                 (short)0, cu, false, false);
    }
    const float wv = ws[16 * c + lm];    // ds_load_b32
#pragma unroll
    for (int r = 0; r < 8; ++r) {        // C tile: row = r + 8*half, col = lm
      accl[r] += __expf(cl[r]) * wv;
      accu[r] += __expf(cu[r]) * wv;
    }
  }

  // Reduce over the 16 column-lanes of each half (xor 1,2,4,8 stays in-half)
#pragma unroll
  for (int r = 0; r < 8; ++r) {
#pragma unroll
    for (int s = 1; s < 16; s <<= 1) {
      accl[r] += __shfl_xor(accl[r], s, 32);
      accu[r] += __shfl_xor(accu[r], s, 32);
    }
  }

  if (lm == 0) {
#pragma unroll
    for (int r = 0; r < 8; ++r) {
      const int orow = rowbase + r + 8 * half;
      const float lo = accl[r];
      const float hi = accu[r];
      out[orow]         = fminf(lo, hi);  // left = min(lo, hi)
      out[BROWS + orow] = hi;             // right = hi
    }
  }
}

// ---------------------------------------------------------------------------
extern "C" void kernel_launch(void* const* d_in, const int* in_sizes, int n_in,
                              void* d_out, int out_size, void* d_ws, size_t ws_size,
                              hipStream_t stream) {
  (void)in_sizes; (void)n_in; (void)out_size; (void)ws_size;
  const float* x  = (const float*)d_in[0];   // (4096, 24) f32
  const float* ie = (const float*)d_in[1];   // (4094, 1)  f32
  float* out  = (float*)d_out;               // left[4096] ++ right[4096]
  float* wpad = (float*)d_ws;                // 4096 f32 scratch

  IE_fm_mobius_kernel<<<dim3(1), dim3(256), 0, stream>>>(ie, wpad);
  IE_choquet_kernel<<<dim3(BROWS / 16), dim3(32), 0, stream>>>(x, wpad, out);
}